// GCN_73830487818377
// MI455X (gfx1250) — compile-verified
//
#include <hip/hip_runtime.h>

// MI455X / gfx1250: wave32, WMMA f32 16x16x4 for the dense h@W stages.
// The workload is edge-scatter bound (~4 GB traffic vs ~8 GFLOP of GEMM),
// so fp32 WMMA keeps full precision at negligible cost.

typedef __attribute__((ext_vector_type(2))) float v2f;
typedef __attribute__((ext_vector_type(8))) float v8f;

#define NFEAT 128

__global__ void zero_kernel(float* __restrict__ p, long n) {
  long i = (long)blockIdx.x * blockDim.x + threadIdx.x;
  if (i < n) p[i] = 0.0f;
}

__global__ void degree_kernel(const int* __restrict__ dst, const float* __restrict__ w,
                              float* __restrict__ deg, int E) {
  int e = blockIdx.x * blockDim.x + threadIdx.x;
  if (e < E) atomicAdd(&deg[dst[e]], w[e]);
}

__global__ void dinv_kernel(float* __restrict__ deg, int N) {
  int i = blockIdx.x * blockDim.x + threadIdx.x;
  if (i < N) {
    float d = deg[i];
    deg[i] = (d > 0.0f) ? rsqrtf(fmaxf(d, 1e-30f)) : 0.0f;
  }
}

__global__ void norm_kernel(const int* __restrict__ src, const int* __restrict__ dst,
                            const float* __restrict__ w, const float* __restrict__ dinv,
                            float* __restrict__ nrm, int E) {
  int e = blockIdx.x * blockDim.x + threadIdx.x;
  if (e < E) nrm[e] = dinv[src[e]] * w[e] * dinv[dst[e]];
}

// C[N x Fout] = A[N x K] * W[K x Fout], fp32 WMMA 16x16x4.
// One wave per 16x16 tile; blockDim.x = 32 * (Fout/16); one 16-row strip per block.
__global__ void gemm_wmma_f32(const float* __restrict__ A, const float* __restrict__ W,
                              float* __restrict__ C, int N, int K, int Fout) {
  const int lane = threadIdx.x & 31;
  const int wcol = threadIdx.x >> 5;        // which 16-col tile this wave owns
  const int row0 = blockIdx.x << 4;         // 16 rows per block
  const int col0 = wcol << 4;
  const int m  = lane & 15;                 // row within tile (A) / col within tile (B,D)
  const int kh = (lane >> 4) << 1;          // K sub-offset: 0 for lanes 0-15, 2 for 16-31

  // clamp A row so EXEC stays all-ones even on a ragged tail tile
  const int arow = (row0 + m < N) ? (row0 + m) : (N - 1);
  const float* __restrict__ Ap = A + (size_t)arow * K;
  const float* __restrict__ Wp = W + col0 + m;   // column (N) index for B

  v8f acc = {};
#pragma unroll 4
  for (int k = 0; k < K; k += 4) {
    v2f a = {}, b = {};
    // A 16x4 f32 layout: lanes 0-15 -> K=k+0,k+1 ; lanes 16-31 -> K=k+2,k+3 (M = lane&15)
    a[0] = Ap[k + kh];
    a[1] = Ap[k + kh + 1];
    // B 4x16 layout: VGPR0 = row K=k+kh, VGPR1 = row K=k+kh+1, N striped over lanes&15
    b[0] = Wp[(size_t)(k + kh) * Fout];
    b[1] = Wp[(size_t)(k + kh + 1) * Fout];
    acc = __builtin_amdgcn_wmma_f32_16x16x4_f32(
        /*neg_a=*/false, a, /*neg_b=*/false, b,
        /*c_mod=*/(short)0, acc, /*reuse_a=*/false, /*reuse_b=*/false);
  }

  // D 16x16 f32 layout: VGPR v -> M = v (lanes 0-15) or 8+v (lanes 16-31); N = lane&15
  const int mb = (lane >> 4) << 3;
#pragma unroll
  for (int v = 0; v < 8; ++v) {
    int row = row0 + mb + v;
    if (row < N) C[(size_t)row * Fout + col0 + m] = acc[v];
  }
}

// out[dst[e]] += norm[e] * h[src[e]] ; one lane per (edge, 4-feature chunk)
__global__ void scatter_kernel(const float* __restrict__ h, const int* __restrict__ src,
                               const int* __restrict__ dst, const float* __restrict__ nrm,
                               float* __restrict__ out, int E, int F, int shift) {
  long tid = (long)blockIdx.x * blockDim.x + threadIdx.x;
  int groups = F >> 2;
  if (tid >= (long)E * groups) return;
  int e = (int)(tid >> shift);
  int g = (int)(tid & (groups - 1));
  float wn = nrm[e];
  int s = src[e];
  int d = dst[e];
  const float4* __restrict__ hv = (const float4*)(h + (size_t)s * F);
  float4 v = hv[g];
  float* o = out + (size_t)d * F + (g << 2);
  atomicAdd(o + 0, wn * v.x);
  atomicAdd(o + 1, wn * v.y);
  atomicAdd(o + 2, wn * v.z);
  atomicAdd(o + 3, wn * v.w);
}

__global__ void bias_relu_kernel(float* __restrict__ h, const float* __restrict__ b,
                                 long total, int fmask) {
  long i = (long)blockIdx.x * blockDim.x + threadIdx.x;
  if (i < total) {
    float v = h[i] + b[i & fmask];
    h[i] = fmaxf(v, 0.0f);
  }
}

// final layer: bias (no relu), duplicate into both tuple outputs
__global__ void bias_out_kernel(const float* __restrict__ h, const float* __restrict__ b,
                                float* __restrict__ out, long total, int fmask) {
  long i = (long)blockIdx.x * blockDim.x + threadIdx.x;
  if (i < total) {
    float v = h[i] + b[i & fmask];
    out[i] = v;
    out[i + total] = v;
  }
}

static inline int cdiv_l(long a, long b) { return (int)((a + b - 1) / b); }

extern "C" void kernel_launch(void* const* d_in, const int* in_sizes, int n_in,
                              void* d_out, int out_size, void* d_ws, size_t ws_size,
                              hipStream_t stream) {
  const float* x  = (const float*)d_in[0];
  const int*   ei = (const int*)d_in[1];
  const float* ew = (const float*)d_in[2];
  const float* W1 = (const float*)d_in[3];
  const float* b1 = (const float*)d_in[4];
  const float* W2 = (const float*)d_in[5];
  const float* b2 = (const float*)d_in[6];
  const float* W3 = (const float*)d_in[7];
  const float* b3 = (const float*)d_in[8];

  const int N = in_sizes[0] / NFEAT;      // 100000
  const int E = in_sizes[2];              // 1600000
  const int F3 = in_sizes[8];             // 64
  const int* src = ei;
  const int* dst = ei + E;

  // workspace carve-up (256B-aligned slices)
  auto align_up = [](size_t v) { return (v + 255) & ~(size_t)255; };
  char* ws = (char*)d_ws;
  float* deg  = (float*)ws; ws += align_up(sizeof(float) * (size_t)N);         // deg -> dinv in place
  float* nrm  = (float*)ws; ws += align_up(sizeof(float) * (size_t)E);
  float* bufA = (float*)ws; ws += align_up(sizeof(float) * (size_t)N * NFEAT);
  float* bufB = (float*)ws;

  const int B = 256;

  // ---- GCN symmetric norm ----
  zero_kernel<<<cdiv_l(N, B), B, 0, stream>>>(deg, N);
  degree_kernel<<<cdiv_l(E, B), B, 0, stream>>>(dst, ew, deg, E);
  dinv_kernel<<<cdiv_l(N, B), B, 0, stream>>>(deg, N);
  norm_kernel<<<cdiv_l(E, B), B, 0, stream>>>(src, dst, ew, deg, nrm, E);

  const int gemmGrid = cdiv_l(N, 16);
  const long nf = (long)N * NFEAT;

  // ---- layer 1: x @ W1 -> bufB ; scatter -> bufA ; +b1, relu ----
  gemm_wmma_f32<<<gemmGrid, 32 * (NFEAT / 16), 0, stream>>>(x, W1, bufB, N, NFEAT, NFEAT);
  zero_kernel<<<cdiv_l(nf, B), B, 0, stream>>>(bufA, nf);
  scatter_kernel<<<cdiv_l((long)E * (NFEAT / 4), B), B, 0, stream>>>(bufB, src, dst, nrm, bufA, E, NFEAT, 5);
  bias_relu_kernel<<<cdiv_l(nf, B), B, 0, stream>>>(bufA, b1, nf, NFEAT - 1);

  // ---- layer 2: bufA @ W2 -> bufB ; scatter -> bufA ; +b2, relu ----
  gemm_wmma_f32<<<gemmGrid, 32 * (NFEAT / 16), 0, stream>>>(bufA, W2, bufB, N, NFEAT, NFEAT);
  zero_kernel<<<cdiv_l(nf, B), B, 0, stream>>>(bufA, nf);
  scatter_kernel<<<cdiv_l((long)E * (NFEAT / 4), B), B, 0, stream>>>(bufB, src, dst, nrm, bufA, E, NFEAT, 5);
  bias_relu_kernel<<<cdiv_l(nf, B), B, 0, stream>>>(bufA, b2, nf, NFEAT - 1);

  // ---- layer 3: bufA @ W3 -> bufB (N x 64) ; scatter -> bufA ; +b3 -> d_out (x2) ----
  const long nf3 = (long)N * F3;
  gemm_wmma_f32<<<gemmGrid, 32 * (F3 / 16), 0, stream>>>(bufA, W3, bufB, N, NFEAT, F3);
  zero_kernel<<<cdiv_l(nf3, B), B, 0, stream>>>(bufA, nf3);
  scatter_kernel<<<cdiv_l((long)E * (F3 / 4), B), B, 0, stream>>>(bufB, src, dst, nrm, bufA, E, F3, 4);
  bias_out_kernel<<<cdiv_l(nf3, B), B, 0, stream>>>(bufA, b3, (float*)d_out, nf3, F3 - 1);
}